// AdaptiveFanInGumbel_57037165691067
// MI455X (gfx1250) — compile-verified
//
#include <hip/hip_runtime.h>
#include <hip/hip_bf16.h>

// Problem constants (from reference): B=8, S=2048, H=2048
#define BB 8
#define SS 2048
#define HH 2048

typedef __attribute__((ext_vector_type(2))) float v2f;
typedef __attribute__((ext_vector_type(8))) float v8f;

// ---------------------------------------------------------------------------
// Kernel 1: fused projection via V_WMMA_F32_16X16X4_F32.
// One wave handles 16 consecutive sequence positions of one batch.
// B-matrix columns: n=0,1 -> W_lo[:,0..1]; n=2,3 -> W_hi[:,0..1].
// Columns n>=4 carry garbage on purpose: B column n only contributes to C
// column n, and we never read C columns >= 4 -- so no zero-select is needed
// (this is what removes the per-iteration saveexec branches from round 1).
// Accumulates over K=HH in steps of 4. Result P01[b*S+s][0..3] =
//   { h_s . Wlo[:,0], h_s . Wlo[:,1], h_s . Whi[:,0], h_s . Whi[:,1] }.
// ---------------------------------------------------------------------------
__global__ void __launch_bounds__(256)
proj_wmma_kernel(const float* __restrict__ hid, const float* __restrict__ W,
                 float* __restrict__ P01) {
  int wave = (int)((blockIdx.x * blockDim.x + threadIdx.x) >> 5);
  int lane = (int)(threadIdx.x & 31);
  int b    = wave / (SS / 16);
  int s0   = (wave % (SS / 16)) * 16;
  int half = lane >> 4;      // K sub-block (ISA A/B layout: lanes 16-31 hold K=2,3)
  int m    = lane & 15;      // A: row M; B: column N

  // A operand: lane holds hid[s0+m][k0 + 2*half + {0,1}]  (16x4 fp32 A layout)
  const float* arow = hid + ((size_t)(b * SS + s0 + m)) * HH + 2 * half;

  // B operand: lane needs W[wrb + k0 + 2*half + {0,1}][wcol].
  // Row-major W[r][c] lives at index 2*r + c. Per-lane base pointer:
  int wcol = m & 1;                    // W output column (0 or 1)
  int wrb  = (m >> 1) ? HH : 0;        // columns 0,1 -> W_lo; others -> W_hi rows
  const float* wpl = W + (size_t)wrb * 2 + wcol + 4 * half;

  v8f c = {};
  #pragma unroll 8
  for (int k0 = 0; k0 < HH; k0 += 4) {
    v2f a = *(const v2f*)(arow + k0);
    v2f bv;
    bv.x = wpl[(size_t)k0 * 2];        // W[wrb + k0 + 2*half    ][wcol]
    bv.y = wpl[(size_t)k0 * 2 + 2];    // W[wrb + k0 + 2*half + 1][wcol]
    c = __builtin_amdgcn_wmma_f32_16x16x4_f32(false, a, false, bv,
                                              (short)0, c, false, false);
  }

  // C/D layout: lane half -> M block of 8, VGPR j -> row, lane%16 -> N column.
  if (m < 4) {
    float* out = P01 + ((size_t)(b * SS + s0 + half * 8)) * 4 + m;
    #pragma unroll
    for (int j = 0; j < 8; ++j) out[(size_t)j * 4] = c[j];
  }
}

// ---------------------------------------------------------------------------
// Kernel 2: logits = P0[s] + P1[s+1] + bias ; hard Gumbel-softmax -> mm.
// ---------------------------------------------------------------------------
__global__ void __launch_bounds__(256)
logits_mm_kernel(const float* __restrict__ P01, const float* __restrict__ bias,
                 const float* __restrict__ gum, const int* __restrict__ amask,
                 const int* __restrict__ smask, float* __restrict__ out_logits,
                 float* __restrict__ out_mm) {
  int i = (int)(blockIdx.x * blockDim.x + threadIdx.x);
  if (i >= BB * SS) return;
  int s = i & (SS - 1);

  float4 p = ((const float4*)P01)[i];
  float l0 = p.x + bias[0];
  float l1 = p.y + bias[1];
  if (s < SS - 1) {                      // pairs at last position use zero stub
    float4 pn = ((const float4*)P01)[i + 1];
    l0 += pn.z;
    l1 += pn.w;
  }
  out_logits[2 * i]     = l0;
  out_logits[2 * i + 1] = l1;

  float g0 = l0 + gum[2 * i], g1 = l1 + gum[2 * i + 1];
  float mx = fmaxf(g0, g1);
  float e0 = __expf(g0 - mx), e1 = __expf(g1 - mx);
  float inv = 1.0f / (e0 + e1);
  float y0 = e0 * inv, y1 = e1 * inv;
  int   idx = (y1 > y0) ? 1 : 0;         // argmax, ties -> 0 (first index)
  float h0 = (idx == 0) ? 1.0f : 0.0f;
  float h1 = (idx == 1) ? 1.0f : 0.0f;
  float mm0 = (h0 - y0) + y0;            // straight-through, same fp order as ref
  float mm1 = (h1 - y1) + y1;
  if (smask[i]) { mm0 = 1.0f; mm1 = 0.0f; }
  float mv = (float)amask[i];
  out_mm[2 * i]     = mm0 * mv;
  out_mm[2 * i + 1] = mm1 * mv;
}

// ---------------------------------------------------------------------------
// Kernel 3: per-batch run-length scan. One block (256 thr x 8 elems) per batch.
// Produces v[s], segment boundaries seg_start[t], new_len, and the small
// outputs (merged mask / special / counts).
// ---------------------------------------------------------------------------
__global__ void __launch_bounds__(256)
scan_kernel(const float* __restrict__ mm, const int* __restrict__ amask,
            float* __restrict__ v_ws, int* __restrict__ seg_ws,
            int* __restrict__ newlen_ws, float* __restrict__ out_mam,
            float* __restrict__ out_msp, float* __restrict__ out_cnt) {
  int b = (int)blockIdx.x;
  int tid = (int)threadIdx.x;

  __shared__ int sh_scan[256];
  __shared__ int sh_T;
  __shared__ int sh_newlen;
  __shared__ int sh_seg[SS + 1];

  for (int t = tid; t < SS + 1; t += 256) sh_seg[t] = 0;
  if (tid == 0) sh_T = 0;
  __syncthreads();

  // T = number of valid tokens
  int tsum = 0;
  #pragma unroll
  for (int j = 0; j < 8; ++j) tsum += amask[b * SS + tid * 8 + j];
  atomicAdd(&sh_T, tsum);
  __syncthreads();
  int T = sh_T;

  // local want/inc/v for this thread's 8 positions
  int s_base = tid * 8;
  int incs[8];
  float vv[8];
  int wprev;
  {
    int sp = s_base - 1;
    if (sp < 0) wprev = 0;
    else {
      float mp = mm[(size_t)(b * SS + sp) * 2 + 1];
      wprev = (mp > 0.5f) && (sp >= 1) && (sp < T - 1);
    }
  }
  int run = 0;
  #pragma unroll
  for (int j = 0; j < 8; ++j) {
    int s = s_base + j;
    float m0 = mm[(size_t)(b * SS + s) * 2 + 0];
    float m1 = mm[(size_t)(b * SS + s) * 2 + 1];
    int  w     = (m1 > 0.5f) && (s >= 1) && (s < T - 1);
    bool valid = (s < T);
    int  inc   = (valid && s >= 1 && !(w && wprev)) ? 1 : 0;
    incs[j] = inc;
    float v = w ? m1 : m0;
    if (s == 0) v = 1.0f;
    if (!valid) v = 0.0f;
    vv[j] = v;
    run += inc;
    wprev = w;
  }

  // Hillis-Steele inclusive scan over 256 per-thread sums
  sh_scan[tid] = run;
  __syncthreads();
  for (int off = 1; off < 256; off <<= 1) {
    int x = sh_scan[tid];
    int y = (tid >= off) ? sh_scan[tid - off] : 0;
    __syncthreads();
    sh_scan[tid] = x + y;
    __syncthreads();
  }
  int acc = sh_scan[tid] - run;   // exclusive prefix

  #pragma unroll
  for (int j = 0; j < 8; ++j) {
    int s = s_base + j;
    acc += incs[j];               // r[s] (inclusive cumsum)
    v_ws[b * SS + s] = vv[j];
    if (incs[j]) sh_seg[acc] = s; // new output row starts at s
    if (s == 0 && T > 0) sh_seg[0] = 0;
    if (s == SS - 1) sh_newlen = acc + 1;
  }
  __syncthreads();
  int NL = sh_newlen;
  if (tid == 0) sh_seg[NL] = T;   // close last segment
  __syncthreads();

  #pragma unroll
  for (int j = 0; j < 8; ++j) {
    int t = s_base + j;
    out_mam[b * SS + t] = (t < NL) ? 1.0f : 0.0f;
    out_msp[b * SS + t] = (t == 0 || t == NL - 1) ? 1.0f : 0.0f;
    out_cnt[b * SS + t] = (t < NL) ? (float)(sh_seg[t + 1] - sh_seg[t]) : 0.0f;
  }
  for (int t = tid; t < SS + 1; t += 256) seg_ws[b * (SS + 1) + t] = sh_seg[t];
  if (tid == 0) newlen_ws[b] = NL;
}

// ---------------------------------------------------------------------------
// Kernel 4: segmented gather-sum (replaces the dense 137-GFLOP einsum).
// One block per output row; bandwidth-optimal: hidden read once, out written
// once (~270 MB total -> ~11.5 us at 23.3 TB/s). Prefetches the next row of
// the current run (global_prefetch_b8) while accumulating.
// ---------------------------------------------------------------------------
__global__ void __launch_bounds__(256)
merge_kernel(const float* __restrict__ hid, const float* __restrict__ v_ws,
             const int* __restrict__ seg_ws, const int* __restrict__ newlen_ws,
             float* __restrict__ out_mh) {
  int b = (int)(blockIdx.x >> 11);       // / SS
  int t = (int)(blockIdx.x & (SS - 1));
  int tid = (int)threadIdx.x;

  float4 a0 = {0.f, 0.f, 0.f, 0.f};
  float4 a1 = {0.f, 0.f, 0.f, 0.f};
  if (t < newlen_ws[b]) {
    int s0 = seg_ws[b * (SS + 1) + t];
    int s1 = seg_ws[b * (SS + 1) + t + 1];
    for (int s = s0; s < s1; ++s) {
      if (s + 1 < s1)
        __builtin_prefetch(hid + ((size_t)(b * SS + s + 1)) * HH + tid * 8, 0, 0);
      float vs = v_ws[b * SS + s];
      const float4* hrow =
          (const float4*)(hid + ((size_t)(b * SS + s)) * HH + tid * 8);
      float4 h0 = hrow[0];
      float4 h1 = hrow[1];
      a0.x += vs * h0.x; a0.y += vs * h0.y; a0.z += vs * h0.z; a0.w += vs * h0.w;
      a1.x += vs * h1.x; a1.y += vs * h1.y; a1.z += vs * h1.z; a1.w += vs * h1.w;
    }
  }
  float4* orow = (float4*)(out_mh + ((size_t)(b * SS + t)) * HH + tid * 8);
  orow[0] = a0;   // rows >= new_len must be zero (d_out is poisoned)
  orow[1] = a1;
}

// ---------------------------------------------------------------------------
extern "C" void kernel_launch(void* const* d_in, const int* in_sizes, int n_in,
                              void* d_out, int out_size, void* d_ws, size_t ws_size,
                              hipStream_t stream) {
  const float* hid   = (const float*)d_in[0];   // [B,S,H] f32
  const int*   amask = (const int*)d_in[1];     // [B,S] i32
  const int*   smask = (const int*)d_in[2];     // [B,S] i32
  const float* gum   = (const float*)d_in[3];   // [B,S,2] f32
  const float* W     = (const float*)d_in[4];   // [2H,2] f32
  const float* bias  = (const float*)d_in[5];   // [2] f32

  // Output layout (flat concat, all f32):
  float* out    = (float*)d_out;
  float* o_mh   = out;                                    // [B,S,H]
  float* o_mam  = o_mh  + (size_t)BB * SS * HH;           // [B,S]
  float* o_msp  = o_mam + (size_t)BB * SS;                // [B,S]
  float* o_cnt  = o_msp + (size_t)BB * SS;                // [B,S]
  float* o_mm   = o_cnt + (size_t)BB * SS;                // [B,S,2]
  float* o_lg   = o_mm  + (size_t)BB * SS * 2;            // [B,S,2]

  // Workspace layout:
  char* ws = (char*)d_ws;
  float* P01    = (float*)ws;                              // [B*S][4]
  ws += (size_t)BB * SS * 4 * sizeof(float);
  float* v_ws   = (float*)ws;                              // [B*S]
  ws += (size_t)BB * SS * sizeof(float);
  int* seg_ws   = (int*)ws;                                // [B][S+1]
  ws += (size_t)BB * (SS + 1) * sizeof(int);
  int* newlen_ws = (int*)ws;                               // [B]

  // K1: projection via WMMA. B*S/16 waves, 8 waves per block.
  int waves = (BB * SS) / 16;
  proj_wmma_kernel<<<waves / 8, 256, 0, stream>>>(hid, W, P01);

  // K2: logits + Gumbel hard softmax.
  logits_mm_kernel<<<(BB * SS) / 256, 256, 0, stream>>>(
      P01, bias, gum, amask, smask, o_lg, o_mm);

  // K3: per-batch run-length scan.
  scan_kernel<<<BB, 256, 0, stream>>>(o_mm, amask, v_ws, seg_ws, newlen_ws,
                                      o_mam, o_msp, o_cnt);

  // K4: segmented merge (one block per output row).
  merge_kernel<<<BB * SS, 256, 0, stream>>>(hid, v_ws, seg_ws, newlen_ws, o_mh);
}